// MixedStateRegularizer_27307402067970
// MI455X (gfx1250) — compile-verified
//
#include <hip/hip_runtime.h>

typedef __attribute__((ext_vector_type(2))) float v2f;
typedef __attribute__((ext_vector_type(8))) float v8f;

#define LATENT 32
#define WAVES_PER_BLOCK 8
#define BLOCK 256

// ---------------------------------------------------------------------------
// One 4-row group: 4 dword loads/lane feed 4 V_WMMA_F32_16X16X4_F32 updating
// all four 16x16 tiles of the 32x32 Gram. A-tile (16x4 of X^T) and B-tile
// (4x16 of X) share the same per-lane layout:
//   lane L, vgpr v  <->  X[r0 + v + 2*(L>=16)][16*blk + (L&15)]
// ---------------------------------------------------------------------------
__device__ __forceinline__ void gram_step(const float* __restrict__ X,
                                          long g, int half, int col,
                                          v8f& c00, v8f& c01, v8f& c10, v8f& c11)
{
    const long r0 = g * 4 + 2 * half;
    const float* __restrict__ row0 = X + r0 * LATENT;
    const float* __restrict__ row1 = row0 + LATENT;

    v2f a0, a1;
    a0.x = row0[col];       a0.y = row1[col];        // column block 0
    a1.x = row0[col + 16];  a1.y = row1[col + 16];   // column block 1

    // (neg_a, A, neg_b, B, c_mod, C, reuse_a, reuse_b)
    c00 = __builtin_amdgcn_wmma_f32_16x16x4_f32(false, a0, false, a0, (short)0, c00, false, false);
    c01 = __builtin_amdgcn_wmma_f32_16x16x4_f32(false, a0, false, a1, (short)0, c01, false, false);
    c10 = __builtin_amdgcn_wmma_f32_16x16x4_f32(false, a1, false, a0, (short)0, c10, false, false);
    c11 = __builtin_amdgcn_wmma_f32_16x16x4_f32(false, a1, false, a1, (short)0, c11, false, false);
}

// ---------------------------------------------------------------------------
// Kernel 1: per-workgroup partial Gram (32x32).
// ---------------------------------------------------------------------------
__global__ __launch_bounds__(BLOCK) void gram_partial_kernel(
    const float* __restrict__ X, float* __restrict__ partial, int ngroups)
{
    __shared__ float acc[WAVES_PER_BLOCK * 1024];

    const int tid  = threadIdx.x;
    const int lane = tid & 31;
    const int w    = tid >> 5;
    const int half = lane >> 4;     // 0: K={0,1}, 1: K={2,3}
    const int col  = lane & 15;     // M / N index within a 16-wide tile

    // wave-uniform loop bounds -> force into SGPRs so the loop is a scalar
    // s_cmp/s_cbranch and EXEC stays all-1s for the WMMAs.
    const int gwave  = __builtin_amdgcn_readfirstlane(blockIdx.x * WAVES_PER_BLOCK + w);
    const int nwaves = gridDim.x * WAVES_PER_BLOCK;

    v8f c00 = {}, c01 = {}, c10 = {}, c11 = {};

    // 2x unrolled stride loop: 8 dword loads in flight per wave.
    int g = gwave;
    for (; g + nwaves < ngroups; g += 2 * nwaves) {
        gram_step(X, (long)g,          half, col, c00, c01, c10, c11);
        gram_step(X, (long)g + nwaves, half, col, c00, c01, c10, c11);
    }
    if (g < ngroups)
        gram_step(X, (long)g, half, col, c00, c01, c10, c11);

    // C/D layout: vgpr r, lane -> (M = r + 8*half [+16*i], N = col [+16*j])
    float* base = acc + w * 1024;
#pragma unroll
    for (int r = 0; r < 8; ++r) {
        const int m = r + 8 * half;
        base[(m)      * 32 + col     ] = c00[r];
        base[(m)      * 32 + col + 16] = c01[r];
        base[(m + 16) * 32 + col     ] = c10[r];
        base[(m + 16) * 32 + col + 16] = c11[r];
    }
    __syncthreads();

    // deterministic cross-wave reduce (fixed order, no float atomics)
#pragma unroll
    for (int e = tid; e < 1024; e += BLOCK) {
        float s = 0.f;
#pragma unroll
        for (int ww = 0; ww < WAVES_PER_BLOCK; ++ww) s += acc[ww * 1024 + e];
        partial[(long)blockIdx.x * 1024 + e] = s;
    }
}

// ---------------------------------------------------------------------------
// Round-robin tournament pairing: 16 disjoint pairs per step, 31 steps cover
// all 496 index pairs of 0..31. Player 0 fixed, others rotate.
// ---------------------------------------------------------------------------
__device__ __forceinline__ int pairP(int k, int step) {
    return (k == 0) ? 0 : ((k - 1 + step) % 31) + 1;
}
__device__ __forceinline__ int pairQ(int k, int step) {
    return ((30 - k + step) % 31) + 1;   // position 31-k (>=16, never 0)
}

// ---------------------------------------------------------------------------
// Kernel 2: reduce partials -> density; purity & trace; parallel cyclic
// Jacobi eigensolver (10 sweeps); entropy; final 4 losses.
// ---------------------------------------------------------------------------
__global__ __launch_bounds__(BLOCK) void finalize_kernel(
    const float* __restrict__ partial, int nwg,
    const float* __restrict__ X, int nrows, int tail_start,
    float* __restrict__ out)
{
    __shared__ float A[32 * 33];       // padded 32x32
    __shared__ float red[BLOCK];
    __shared__ float cvals[16], svals[16];
    __shared__ float s_purity, s_trace;

    const int t = threadIdx.x;
    const float inv = 1.0f / (float)nrows;

    // density = (sum of partials + scalar tail rows) / n
#pragma unroll
    for (int e = t; e < 1024; e += BLOCK) {
        float s = 0.f;
        for (int wg = 0; wg < nwg; ++wg) s += partial[(long)wg * 1024 + e];
        const int i = e >> 5, j = e & 31;
        for (int r = tail_start; r < nrows; ++r)
            s += X[(long)r * LATENT + i] * X[(long)r * LATENT + j];
        A[i * 33 + j] = s * inv;
    }
    __syncthreads();

    // purity = sum(rho * rho^T); trace
    float p = 0.f;
#pragma unroll
    for (int e = t; e < 1024; e += BLOCK) {
        const int i = e >> 5, j = e & 31;
        p += A[i * 33 + j] * A[j * 33 + i];
    }
    red[t] = p;
    __syncthreads();
    for (int off = BLOCK / 2; off > 0; off >>= 1) {
        if (t < off) red[t] += red[t + off];
        __syncthreads();
    }
    if (t == 0) {
        s_purity = red[0];
        float tr = 0.f;
        for (int i = 0; i < 32; ++i) tr += A[i * 33 + i];
        s_trace = tr;
    }
    __syncthreads();

    // parallel cyclic Jacobi: A <- J^T A J with 16 disjoint Givens/step
    for (int sweep = 0; sweep < 10; ++sweep) {
        for (int step = 0; step < 31; ++step) {
            if (t < 16) {
                const int pp = pairP(t, step), qq = pairQ(t, step);
                const float app = A[pp * 33 + pp];
                const float aqq = A[qq * 33 + qq];
                const float apq = A[pp * 33 + qq];
                float cc = 1.f, ss = 0.f;
                if (fabsf(apq) > 1e-12f) {
                    const float theta = (aqq - app) / (2.f * apq);
                    const float tt = copysignf(1.f, theta) /
                                     (fabsf(theta) + sqrtf(theta * theta + 1.f));
                    cc = 1.f / sqrtf(tt * tt + 1.f);
                    ss = tt * cc;
                }
                cvals[t] = cc; svals[t] = ss;
            }
            __syncthreads();
            // row pass: B = J^T A  (512 disjoint tasks)
            for (int task = t; task < 512; task += BLOCK) {
                const int k = task >> 5, c = task & 31;
                const int pp = pairP(k, step), qq = pairQ(k, step);
                const float cc = cvals[k], ss = svals[k];
                const float ap = A[pp * 33 + c], aq = A[qq * 33 + c];
                A[pp * 33 + c] = cc * ap - ss * aq;
                A[qq * 33 + c] = ss * ap + cc * aq;
            }
            __syncthreads();
            // column pass: A = B J
            for (int task = t; task < 512; task += BLOCK) {
                const int k = task >> 5, r = task & 31;
                const int pp = pairP(k, step), qq = pairQ(k, step);
                const float cc = cvals[k], ss = svals[k];
                const float ap = A[r * 33 + pp], aq = A[r * 33 + qq];
                A[r * 33 + pp] = cc * ap - ss * aq;
                A[r * 33 + qq] = ss * ap + cc * aq;
            }
            __syncthreads();
        }
    }

    if (t == 0) {
        float ent = 0.f;
        for (int i = 0; i < 32; ++i) {
            float lam = A[i * 33 + i];
            lam = fmaxf(lam, 1e-10f);
            ent -= lam * log2f(lam + 1e-10f);
        }
        const float el = fabsf(ent - 5.0f);          // log2(32) = 5
        const float pl = fabsf(s_purity - 0.5f);
        const float tl = fabsf(s_trace - 1.0f);
        out[0] = 0.05f * el + 0.05f * pl + 0.01f * tl;
        out[1] = el;
        out[2] = pl;
        out[3] = tl;
    }
}

// ---------------------------------------------------------------------------
extern "C" void kernel_launch(void* const* d_in, const int* in_sizes, int n_in,
                              void* d_out, int out_size, void* d_ws, size_t ws_size,
                              hipStream_t stream)
{
    (void)n_in; (void)out_size;
    const float* X = (const float*)d_in[0];
    const int nrows = in_sizes[0] / LATENT;
    float* out = (float*)d_out;
    float* partial = (float*)d_ws;

    int nwg = (int)(ws_size / (1024 * sizeof(float)));
    if (nwg > 256) nwg = 256;
    if (nwg < 1)  nwg = 1;

    const int ngroups    = nrows / 4;
    const int tail_start = ngroups * 4;

    gram_partial_kernel<<<nwg, BLOCK, 0, stream>>>(X, partial, ngroups);
    finalize_kernel<<<1, BLOCK, 0, stream>>>(partial, nwg, X, nrows, tail_start, out);
}